// Softmax_79302276153651
// MI455X (gfx1250) — compile-verified
//
#include <hip/hip_runtime.h>

// Softmax over [2048, 32000] f32, row-normalized.
// One workgroup per row; row (128 KB) staged in LDS (320 KB/WGP on CDNA5) via
// gfx1250 async global->LDS DMA (ASYNCcnt) so HBM is touched exactly once per
// element: 262 MB read + 262 MB write = memory-roofline floor (~22.5 us @ 23.3 TB/s).

#define COLS    32000
#define CHUNKS  (COLS / 4)                       // 8000 float4 chunks per row
#define BLOCK   256                              // 8 waves (wave32)
#define WAVES   (BLOCK / 32)
#define ITERS   ((CHUNKS + BLOCK - 1) / BLOCK)   // 32 (31 full + 64-chunk tail)
#define FULL_ITERS 31                            // iterations with all 256 threads active
#define CHUNK_STRIDE_B (BLOCK * 16)              // 4096 bytes per iteration step

typedef __attribute__((ext_vector_type(4))) float v4f;

// One async 16B/lane copy. The 24-bit instruction offset is applied to BOTH the
// LDS destination and the global source (CDNA5 ISA), so a single
// base LDS address VGPR + single global address VGPR pair serves all chunks.
template <int BYTE_OFF>
__device__ __forceinline__ void async_ld_b128(unsigned lds_addr, const v4f* gaddr) {
    asm volatile("global_load_async_to_lds_b128 %0, %1, off offset:%c2"
                 :: "v"(lds_addr), "v"(gaddr), "i"(BYTE_OFF)
                 : "memory");
}

template <int I, int N>
struct IssueLoads {
    static __device__ __forceinline__ void run(unsigned lds_addr, const v4f* gaddr) {
        async_ld_b128<I * CHUNK_STRIDE_B>(lds_addr, gaddr);
        IssueLoads<I + 1, N>::run(lds_addr, gaddr);
    }
};
template <int N>
struct IssueLoads<N, N> {
    static __device__ __forceinline__ void run(unsigned, const v4f*) {}
};

extern "C" __global__ __launch_bounds__(BLOCK)
void softmax_row_kernel(const float* __restrict__ in, float* __restrict__ out) {
    extern __shared__ float smem[];              // COLS row buffer + reduction slots
    float* red = smem + COLS;

    const int tid  = (int)threadIdx.x;
    const int lane = tid & 31;
    const int wave = tid >> 5;
    const size_t row_base = (size_t)blockIdx.x * (size_t)COLS;
    v4f* lrow = (v4f*)smem;
    const v4f* grow = (const v4f*)(in + row_base);

    // ---- Stage 1: issue the whole-row DMA (global -> LDS, ASYNCcnt-tracked) ----
    // Flat shared-aperture pointers carry the LDS byte offset in addr[31:0].
    const unsigned l0 = (unsigned)(size_t)(lrow + tid);
    const v4f* g0 = grow + tid;
    IssueLoads<0, FULL_ITERS>::run(l0, g0);      // 31 x b128 per thread
    if (tid < (CHUNKS - FULL_ITERS * BLOCK))     // 64-chunk tail (waves 0-1 only)
        async_ld_b128<FULL_ITERS * CHUNK_STRIDE_B>(l0, g0);

    // ---- Stage 2: exp in-place + partial sum, pipelined against the DMA ----
    // Each thread consumes only chunks it issued itself, and async loads return
    // done IN ORDER per wave -> per-wave s_wait_asynccnt suffices (no barrier).
    // Waves 2-7 issued 31 loads, waves 0-1 issued 32: cnt<=16 guarantees the
    // first 15 loads of every wave are complete.
    float psum = 0.0f;
    asm volatile("s_wait_asynccnt 16" ::: "memory");
    #pragma unroll 4
    for (int i = 0; i < 15; ++i) {
        const int c = i * BLOCK + tid;
        v4f v = lrow[c];
        v4f e;
        e.x = __expf(v.x); e.y = __expf(v.y);
        e.z = __expf(v.z); e.w = __expf(v.w);
        lrow[c] = e;
        psum += (e.x + e.y) + (e.z + e.w);
    }
    asm volatile("s_wait_asynccnt 0" ::: "memory");
    #pragma unroll 4
    for (int i = 15; i < ITERS; ++i) {
        const int c = i * BLOCK + tid;
        if (c < CHUNKS) {
            v4f v = lrow[c];
            v4f e;
            e.x = __expf(v.x); e.y = __expf(v.y);
            e.z = __expf(v.z); e.w = __expf(v.w);
            lrow[c] = e;
            psum += (e.x + e.y) + (e.z + e.w);
        }
    }

    // ---- wave32 butterfly reduction, then cross-wave via LDS ----
    #pragma unroll
    for (int off = 16; off > 0; off >>= 1)
        psum += __shfl_xor(psum, off, 32);
    if (lane == 0) red[wave] = psum;
    __syncthreads();
    if (tid == 0) {
        float s = 0.0f;
        #pragma unroll
        for (int w = 0; w < WAVES; ++w) s += red[w];
        red[WAVES] = 1.0f / s;
    }
    __syncthreads();
    const float inv = red[WAVES];

    // ---- Stage 3: scale from LDS and stream out (b128 stores) ----
    v4f* orow = (v4f*)(out + row_base);
    #pragma unroll 4
    for (int i = 0; i < ITERS; ++i) {
        const int c = i * BLOCK + tid;
        if (c < CHUNKS)
            orow[c] = lrow[c] * inv;
    }
}

extern "C" void kernel_launch(void* const* d_in, const int* in_sizes, int n_in,
                              void* d_out, int out_size, void* d_ws, size_t ws_size,
                              hipStream_t stream) {
    (void)n_in; (void)out_size; (void)d_ws; (void)ws_size;
    const float* in = (const float*)d_in[0];
    float* out = (float*)d_out;
    const int rows = in_sizes[0] / COLS;         // 2048
    const size_t shmem = (size_t)COLS * sizeof(float) + (WAVES + 2) * sizeof(float);
    softmax_row_kernel<<<dim3(rows), dim3(BLOCK), shmem, stream>>>(in, out);
}